// CenterLossForward_72353019068995
// MI455X (gfx1250) — compile-verified
//
#include <hip/hip_runtime.h>

// Center loss (forward + sparse center update) for MI455X / gfx1250.
//
// Outputs (flat, in return order): d_out[0] = loss, d_out[1..] = new_centers.
//
// Structure (bandwidth-bound; ~424 MB total traffic => ~18us @ 23.3 TB/s):
//   K1: stream-copy centers -> d_out+1 (200MB R + 200MB W), zero loss slot.
//   K2: one wave per 16 samples. Gather center rows, delta = batch - center,
//       scatter-add ALPHA*delta with global_atomic_add_f32 (handles duplicate
//       class indices), and accumulate the Gram matrix Delta*Delta^T with
//       V_WMMA_F32_16X16X4_F32 -- its diagonal gives the per-row squared
//       norms for the loss, in full f32 precision.

#define NUM_CLASSES 100000
#define EMBED       512
#define BATCH       4096
#define LAMBDA      0.01f
#define ALPHA       0.1f

typedef float v2f __attribute__((ext_vector_type(2)));
typedef float v8f __attribute__((ext_vector_type(8)));

// ---------------------------------------------------------------------------
// Kernel 1: new_centers = centers (grid-stride streaming copy) + zero loss.
// centers is 16B-aligned -> b128 loads; d_out+1 is only 4B-aligned -> dword
// stores (still fully coalesced).
// ---------------------------------------------------------------------------
__global__ void __launch_bounds__(256)
centers_copy_kernel(const float* __restrict__ centers, float* __restrict__ out) {
    if (blockIdx.x == 0 && threadIdx.x == 0) out[0] = 0.0f;  // loss accumulator
    float* __restrict__ dst = out + 1;

    const long long total4 = (long long)NUM_CLASSES * EMBED / 4;  // 12.8M float4
    const float4* __restrict__ src = (const float4*)centers;

    long long i      = (long long)blockIdx.x * blockDim.x + threadIdx.x;
    long long stride = (long long)gridDim.x * blockDim.x;
    for (; i < total4; i += stride) {
        float4 v = src[i];              // global_load_b128 (aligned)
        float* p = dst + i * 4;         // 4B-aligned destination
        p[0] = v.x; p[1] = v.y; p[2] = v.z; p[3] = v.w;
    }
}

// ---------------------------------------------------------------------------
// Kernel 2: per-sample delta, scatter-add, and WMMA-based loss reduction.
//
// Wave-to-data mapping for V_WMMA_F32_16X16X4_F32 (wave32):
//   A (16x4 f32):  lane m in 0..15 holds {K=k0, k0+1}, lanes 16..31 hold
//                  {K=k0+2, k0+3} for the same rows (2 VGPRs).
//   B (4x16 f32):  identical striping with N on lanes -> the SAME registers
//                  act as B = A^T, so wmma(a, a, acc) accumulates Delta*Delta^T.
//   C/D (16x16 f32, 8 VGPRs): element (M,N): lanes 0-15 -> (r, lane),
//                  lanes 16-31 -> (r+8, lane-16). Diagonal (i,i):
//                  i<8  -> lane i,     VGPR i
//                  i>=8 -> lane i+16,  VGPR i-8
// ---------------------------------------------------------------------------
__global__ void __launch_bounds__(256)
center_loss_kernel(const int* __restrict__ y,
                   const float* __restrict__ batch,
                   const float* __restrict__ centers,
                   float* __restrict__ out) {
    float* __restrict__ loss = out;
    float* __restrict__ newc = out + 1;

    const int tid  = blockIdx.x * blockDim.x + threadIdx.x;
    const int wave = tid >> 5;            // wave32
    const int lane = threadIdx.x & 31;

    const int row   = wave * 16 + (lane & 15);   // sample handled by this lane
    const int kpair = (lane >> 4) << 1;          // 0 for lanes 0-15, 2 for 16-31

    const int cls = y[row];
    const float* __restrict__ brow = batch   + (long long)row * EMBED;
    const float* __restrict__ crow = centers + (long long)cls * EMBED;
    float*       __restrict__ nrow = newc    + (long long)cls * EMBED;

    v8f acc = {};  // Gram-matrix accumulator (16x16 f32 across the wave)

    for (int k = 0; k < EMBED; k += 4) {
        const int col = k + kpair;                       // even -> 8B aligned
        v2f bv = *(const v2f*)(brow + col);              // global_load_b64
        v2f cv = *(const v2f*)(crow + col);              // gathered center row
        v2f a  = bv - cv;                                // delta pair

        // acc += Delta_tile * Delta_tile^T   (f32 WMMA, K=4 per step)
        acc = __builtin_amdgcn_wmma_f32_16x16x4_f32(
                  /*neg_a=*/false, a, /*neg_b=*/false, a,
                  /*c_mod=*/(short)0, acc,
                  /*reuse_a=*/false, /*reuse_b=*/false);

        // Sparse scatter-add into new_centers; atomics make duplicate class
        // indices accumulate (torch sparse-add semantics).
        atomicAdd(nrow + col,     ALPHA * a.x);          // global_atomic_add_f32
        atomicAdd(nrow + col + 1, ALPHA * a.y);
    }

    // Extract the Gram diagonal (one row-norm per owning lane) and fold into
    // the loss: loss = (LAMBDA / BATCH) * sum_i ||delta_i||^2.
    const int want = (lane < 8) ? lane : ((lane >= 24) ? (lane - 24) : -1);
    float diag = 0.0f;
#pragma unroll
    for (int j = 0; j < 8; ++j)
        if (j == want) diag = acc[j];
    if (want >= 0)
        atomicAdd(loss, (LAMBDA / (float)BATCH) * diag);
}

// ---------------------------------------------------------------------------
extern "C" void kernel_launch(void* const* d_in, const int* in_sizes, int n_in,
                              void* d_out, int out_size, void* d_ws, size_t ws_size,
                              hipStream_t stream) {
    const int*   y       = (const int*)d_in[0];     // [BATCH] int32
    const float* batch   = (const float*)d_in[1];   // [BATCH, EMBED] f32
    const float* centers = (const float*)d_in[2];   // [NUM_CLASSES, EMBED] f32
    float*       out     = (float*)d_out;           // [1 + NUM_CLASSES*EMBED]

    // K1: copy table + zero loss. Must fully precede K2 (same stream => ordered).
    centers_copy_kernel<<<2048, 256, 0, stream>>>(centers, out);

    // K2: 16 samples per wave, 8 waves per 256-thread block -> 128 samples/block.
    center_loss_kernel<<<BATCH / 128, 256, 0, stream>>>(y, batch, centers, out);
}